// relative_depth_crit_77567109366401
// MI455X (gfx1250) — compile-verified
//
#include <hip/hip_runtime.h>
#include <math.h>

// Problem constants (match reference)
#define BB 64
#define PP 50000
#define HH 512
#define WW 512
#define NPTS (BB * PP)          // 3,200,000 points
#define NV4  (NPTS / 4)         // 800,000 vec4 groups (4 | P -> groups never cross batch)
#define BLOCK 256
#define GRID1 1024
#define CHUNKS (NV4 / BLOCK)    // 3125 chunks, exact (3125*256 == 800000)

typedef int   v4i __attribute__((ext_vector_type(4)));
typedef float v4f __attribute__((ext_vector_type(4)));

// ---- CDNA5 async global->LDS path (gfx1250), with compile-safe fallback ----
#if __has_builtin(__builtin_amdgcn_global_load_async_to_lds_b128) && \
    __has_builtin(__builtin_amdgcn_s_wait_asynccnt)
#define USE_ASYNC_LDS 1
#pragma message("CDNA5: using global_load_async_to_lds_b128 + s_wait_asynccnt path")
#else
#define USE_ASYNC_LDS 0
#pragma message("CDNA5: async-to-LDS builtins not found; direct NT-load fallback")
#endif

#if USE_ASYNC_LDS
// Builtin signature (from clang diagnostic): first param is a pointer to
// <4 x i32> in the global address space, second in LDS address space.
typedef __attribute__((address_space(1))) v4i* g_v4i_ptr;
typedef __attribute__((address_space(3))) v4i* l_v4i_ptr;

__device__ __forceinline__ void async_b128(const void* g, void* l) {
    __builtin_amdgcn_global_load_async_to_lds_b128(
        (g_v4i_ptr)g, (l_v4i_ptr)l, /*imm offset*/ 0, /*cpol*/ 0);
}
#endif

// Per-point math for a vec4 group: two L2-resident random gathers + logistic loss.
__device__ __forceinline__ float point_loss4(const float* __restrict__ img,
                                             v4i ax, v4i ay, v4i bx, v4i by, v4f g) {
    float acc = 0.0f;
#pragma unroll
    for (int k = 0; k < 4; ++k) {
        float zA = img[ay[k] * WW + ax[k]];
        float zB = img[by[k] * WW + bx[k]];
        float d  = zA - zB;
        float gv = g[k];
        float m  = fabsf(gv);                 // gt in {-1,0,1} -> mask
        float sp = log1pf(__expf(-gv * d));   // bounded: |gv*d| is small
        acc += m * sp + (1.0f - m) * d * d;
    }
    return acc;
}

__global__ __launch_bounds__(BLOCK) void rd_partial_kernel(
    const float* __restrict__ x,
    const int*   __restrict__ xA, const int* __restrict__ yA,
    const int*   __restrict__ xB, const int* __restrict__ yB,
    const float* __restrict__ gt,
    float* __restrict__ partials)
{
    __shared__ float wave_sums[BLOCK / 32];

    const int t = threadIdx.x;

    const v4i* __restrict__ vxA = (const v4i*)xA;
    const v4i* __restrict__ vyA = (const v4i*)yA;
    const v4i* __restrict__ vxB = (const v4i*)xB;
    const v4i* __restrict__ vyB = (const v4i*)yB;
    const v4f* __restrict__ vgt = (const v4f*)gt;

    float acc = 0.0f;

#if USE_ASYNC_LDS
    // Double-buffered async staging of the 5 streams: 2 x 5 x 256 x 16B = 40 KB LDS.
    // Each thread owns slot [.][.][t]; no cross-thread LDS sharing -> no barriers,
    // ordering purely via ASYNCcnt (in-order completion).
    __shared__ v4i stage[2][5][BLOCK];

    auto issue_chunk = [&](int c, int nb) {
        const int i = c * BLOCK + t;          // vec4 group index
        async_b128(vxA + i, &stage[nb][0][t]);
        async_b128(vyA + i, &stage[nb][1][t]);
        async_b128(vxB + i, &stage[nb][2][t]);
        async_b128(vyB + i, &stage[nb][3][t]);
        async_b128(vgt + i, &stage[nb][4][t]);
    };

    int c  = blockIdx.x;
    int nb = 0;
    if (c < CHUNKS) issue_chunk(c, 0);        // prologue: prime buffer 0

    for (; c < CHUNKS; c += GRID1) {
        const int cn = c + GRID1;
        if (cn < CHUNKS) {
            issue_chunk(cn, nb ^ 1);          // overlap next chunk's streaming fetch
            __builtin_amdgcn_s_wait_asynccnt(5);  // only the 5 just-issued may remain
        } else {
            __builtin_amdgcn_s_wait_asynccnt(0);  // drain last chunk
        }

        const int i = c * BLOCK + t;
        v4i ax = stage[nb][0][t];
        v4i ay = stage[nb][1][t];
        v4i bx = stage[nb][2][t];
        v4i by = stage[nb][3][t];
        v4f g  = __builtin_bit_cast(v4f, stage[nb][4][t]);

        const int b = (i * 4) / PP;           // same batch for all 4 points (4 | P)
        const float* __restrict__ img = x + (size_t)b * (size_t)(HH * WW);
        acc += point_loss4(img, ax, ay, bx, by, g);
        nb ^= 1;
    }
#else
    // Fallback: direct 128-bit non-temporal loads + software prefetch.
    const int tid      = blockIdx.x * BLOCK + t;
    const int nthreads = GRID1 * BLOCK;
    for (int i = tid; i < NV4; i += nthreads) {
        int inext = i + nthreads;
        if (inext < NV4) {
            __builtin_prefetch((const void*)(vxA + inext), 0, 1);
            __builtin_prefetch((const void*)(vyA + inext), 0, 1);
            __builtin_prefetch((const void*)(vxB + inext), 0, 1);
            __builtin_prefetch((const void*)(vyB + inext), 0, 1);
            __builtin_prefetch((const void*)(vgt + inext), 0, 1);
        }
        v4i ax = __builtin_nontemporal_load(vxA + i);
        v4i ay = __builtin_nontemporal_load(vyA + i);
        v4i bx = __builtin_nontemporal_load(vxB + i);
        v4i by = __builtin_nontemporal_load(vyB + i);
        v4f g  = __builtin_nontemporal_load(vgt + i);

        const int b = (i * 4) / PP;
        const float* __restrict__ img = x + (size_t)b * (size_t)(HH * WW);
        acc += point_loss4(img, ax, ay, bx, by, g);
    }
#endif

    // wave32 tree reduce, fixed order (deterministic)
#pragma unroll
    for (int off = 16; off > 0; off >>= 1)
        acc += __shfl_down(acc, off, 32);

    const int lane = t & 31;
    const int wave = t >> 5;
    if (lane == 0) wave_sums[wave] = acc;
    __syncthreads();

    if (t == 0) {
        float s = 0.0f;
#pragma unroll
        for (int w = 0; w < BLOCK / 32; ++w) s += wave_sums[w];
        partials[blockIdx.x] = s;             // unconditionally written every call
    }
}

// Kernel 2: single block reduces GRID1 partials in fixed pairing order; writes mean.
__global__ __launch_bounds__(256) void rd_final_kernel(
    const float* __restrict__ partials, float* __restrict__ out)
{
    __shared__ float sm[256];
    float s = 0.0f;
    for (int i = threadIdx.x; i < GRID1; i += 256)
        s += partials[i];
    sm[threadIdx.x] = s;
    __syncthreads();
#pragma unroll
    for (int stride = 128; stride > 0; stride >>= 1) {
        if ((int)threadIdx.x < stride) sm[threadIdx.x] += sm[threadIdx.x + stride];
        __syncthreads();
    }
    if (threadIdx.x == 0)
        out[0] = sm[0] * (1.0f / (float)NPTS);
}

extern "C" void kernel_launch(void* const* d_in, const int* in_sizes, int n_in,
                              void* d_out, int out_size, void* d_ws, size_t ws_size,
                              hipStream_t stream) {
    // setup_inputs() order: x, x_A, y_A, x_B, y_B, gt
    const float* x  = (const float*)d_in[0];
    const int*   xA = (const int*)  d_in[1];
    const int*   yA = (const int*)  d_in[2];
    const int*   xB = (const int*)  d_in[3];
    const int*   yB = (const int*)  d_in[4];
    const float* gt = (const float*)d_in[5];

    float* partials = (float*)d_ws;     // GRID1 floats = 4 KB scratch
    float* out      = (float*)d_out;    // 1 float (scalar mean)

    (void)in_sizes; (void)n_in; (void)out_size; (void)ws_size;

    rd_partial_kernel<<<GRID1, BLOCK, 0, stream>>>(x, xA, yA, xB, yB, gt, partials);
    rd_final_kernel<<<1, 256, 0, stream>>>(partials, out);
}